// EnhancedLoss_53549652246844
// MI455X (gfx1250) — compile-verified
//
#include <hip/hip_runtime.h>
#include <hip/hip_bf16.h>

#define ALPHA   0.5f
#define BETA    0.5f
#define GAMMA   1.0f
#define SMOOTH  1e-5f

#define BATCH 16
#define HH    512
#define WW    512
#define HW    (HH * WW)        // 262144 = 2^18
#define NTOT  (BATCH * HW)     // 4194304

#define LOG2E 1.44269504088896340736f
#define LN2   0.69314718055994530942f

typedef __attribute__((ext_vector_type(2))) float v2f;
typedef __attribute__((ext_vector_type(4))) float v4f;
typedef __attribute__((ext_vector_type(8))) float v8f;

// ---------------- fused reduction with V_WMMA_F32_16X16X4_F32 ----------------
#define RED_BLOCKS  512
#define RED_THREADS 256
#define RED_WAVES   (RED_BLOCKS * RED_THREADS / 32)   // 4096
#define RED_EPI     128                               // elements / wave / iter
#define RED_STRIDE  (RED_WAVES * RED_EPI)             // 524288
#define RED_ITERS   (NTOT / RED_STRIDE)               // 8

__device__ __forceinline__ v8f wmma_acc(v2f a, v2f ones, v8f c) {
    return __builtin_amdgcn_wmma_f32_16x16x4_f32(false, a, false, ones, (short)0,
                                                 c, false, false);
}

__global__ void loss_reduce_kernel(const float* __restrict__ xin,
                                   const float* __restrict__ tin,
                                   float* __restrict__ partials) {
    const int wave = (blockIdx.x * blockDim.x + threadIdx.x) >> 5;
    const int lane = threadIdx.x & 31;

    v8f cP = {}; // sum(probs)
    v8f cI = {}; // sum(probs * t)
    v8f cT = {}; // sum(t)
    v8f cE = {}; // sum(softplus(x) - x*t)
    const v2f ones = {1.0f, 1.0f};

    const int base = wave * RED_EPI + lane * 4;
    for (int it = 0; it < RED_ITERS; ++it) {
        const int e = base + it * RED_STRIDE;
        v4f xv = *(const v4f*)(xin + e);               // global_load_b128
        v4f tv = *(const v4f*)(tin + e);
        // speculative prefetch of next iteration's stream (global_prefetch_b8)
        __builtin_prefetch(xin + e + RED_STRIDE, 0, 1);
        __builtin_prefetch(tin + e + RED_STRIDE, 0, 1);

        v4f pv, iv, ev;
#pragma unroll
        for (int k = 0; k < 4; ++k) {
            float x  = xv[k];
            float t  = tv[k];
            // e = exp(-|x|) in (0,1] : raw v_exp_f32 (no denorm/inf possible here)
            float ea = __builtin_amdgcn_exp2f(-fabsf(x) * LOG2E);
            float op = 1.0f + ea;                      // in (1,2] -> raw log safe
            float r  = __builtin_amdgcn_rcpf(op);
            float pr = (x >= 0.0f) ? r : ea * r;       // sigmoid(x)
            pv[k] = pr;
            iv[k] = pr * t;
            // softplus(x) - x*t = max(x,0) + log(1+e) - x*t
            ev[k] = fmaxf(x, 0.0f) - x * t + __builtin_amdgcn_logf(op) * LN2;
        }
        v2f a0, a1;
        a0[0] = pv[0]; a0[1] = pv[1]; a1[0] = pv[2]; a1[1] = pv[3];
        cP = wmma_acc(a0, ones, cP); cP = wmma_acc(a1, ones, cP);
        a0[0] = iv[0]; a0[1] = iv[1]; a1[0] = iv[2]; a1[1] = iv[3];
        cI = wmma_acc(a0, ones, cI); cI = wmma_acc(a1, ones, cI);
        a0[0] = tv[0]; a0[1] = tv[1]; a1[0] = tv[2]; a1[1] = tv[3];
        cT = wmma_acc(a0, ones, cT); cT = wmma_acc(a1, ones, cT);
        a0[0] = ev[0]; a0[1] = ev[1]; a1[0] = ev[2]; a1[1] = ev[3];
        cE = wmma_acc(a0, ones, cE); cE = wmma_acc(a1, ones, cE);
    }

    // Column j of D is the full 16-row sum vector; column 0 lives in lane 0
    // (rows 0-7 in VGPR 0-7) and lane 16 (rows 8-15).
    float sP = 0.f, sI = 0.f, sT = 0.f, sE = 0.f;
#pragma unroll
    for (int k = 0; k < 8; ++k) { sP += cP[k]; sI += cI[k]; sT += cT[k]; sE += cE[k]; }

    float tPs = __shfl(sP, 0, 32) + __shfl(sP, 16, 32);
    float tIs = __shfl(sI, 0, 32) + __shfl(sI, 16, 32);
    float tTs = __shfl(sT, 0, 32) + __shfl(sT, 16, 32);
    float tEs = __shfl(sE, 0, 32) + __shfl(sE, 16, 32);

    if (lane == 0) {
        partials[wave * 4 + 0] = tPs;
        partials[wave * 4 + 1] = tIs;
        partials[wave * 4 + 2] = tTs;
        partials[wave * 4 + 3] = tEs;
    }
}

// ---------------- connected components (union-find, 8-connectivity) ----------------
__global__ void zero_counts_kernel(int* __restrict__ counts) {
    if (threadIdx.x < 32) counts[threadIdx.x] = 0;
}

__global__ void ccl_init_kernel(const float* __restrict__ src, float thresh,
                                int* __restrict__ labels) {
    int i = blockIdx.x * blockDim.x + threadIdx.x;
    if (i < NTOT) labels[i] = (src[i] > thresh) ? i : -1;
}

__device__ __forceinline__ int ccl_find(int* L, int i) {
    int p = L[i];
    while (p != i) { i = p; p = L[i]; }   // labels only decrease -> terminates
    return i;
}

__device__ __forceinline__ void ccl_union(int* L, int a, int b) {
    for (;;) {
        a = ccl_find(L, a);
        b = ccl_find(L, b);
        if (a == b) return;
        int lo = a < b ? a : b;
        int hi = a ^ b ^ lo;
        int old = atomicCAS(&L[hi], hi, lo);
        if (old == hi) return;
        a = old; b = lo;                   // root moved under us; retry
    }
}

__global__ void ccl_merge_kernel(int* __restrict__ labels) {
    int i = blockIdx.x * blockDim.x + threadIdx.x;
    if (i >= NTOT) return;
    if (labels[i] < 0) return;             // background never changes
    int local = i & (HW - 1);
    int y = local >> 9;
    int x = local & (WW - 1);
    if (x > 0 && labels[i - 1] >= 0) ccl_union(labels, i, i - 1);            // W
    if (y > 0) {
        int up = i - WW;
        if (labels[up] >= 0) ccl_union(labels, i, up);                       // N
        if (x > 0      && labels[up - 1] >= 0) ccl_union(labels, i, up - 1); // NW
        if (x < WW - 1 && labels[up + 1] >= 0) ccl_union(labels, i, up + 1); // NE
    }
}

__global__ void ccl_count_kernel(const int* __restrict__ labels,
                                 int* __restrict__ counts) {
    int i = blockIdx.x * blockDim.x + threadIdx.x;
    if (i < NTOT && labels[i] == i)        // exactly one root per component
        atomicAdd(&counts[i >> 18], 1);    // HW == 2^18
}

// ---------------- deterministic finalize ----------------
__global__ void finalize_kernel(const float* __restrict__ partials,
                                const int* __restrict__ cnt_pred,
                                const int* __restrict__ cnt_tgt,
                                float* __restrict__ out) {
    int lane = threadIdx.x;                // one wave32
    float sP = 0.f, sI = 0.f, sT = 0.f, sE = 0.f;
    for (int w = lane; w < RED_WAVES; w += 32) {
        sP += partials[w * 4 + 0];
        sI += partials[w * 4 + 1];
        sT += partials[w * 4 + 2];
        sE += partials[w * 4 + 3];
    }
#pragma unroll
    for (int off = 16; off > 0; off >>= 1) {   // fixed tree -> deterministic
        sP += __shfl_xor(sP, off, 32);
        sI += __shfl_xor(sI, off, 32);
        sT += __shfl_xor(sT, off, 32);
        sE += __shfl_xor(sE, off, 32);
    }
    if (lane == 0) {
        float dice = 1.0f - (2.0f * sI + SMOOTH) / (sP + sT + SMOOTH);
        float ce   = sE / (float)NTOT;
        float region = 0.0f;
        for (int b = 0; b < BATCH; ++b)
            region += fabsf((float)(cnt_pred[b] - cnt_tgt[b]));
        region /= (float)BATCH;
        out[0] = ALPHA * dice + BETA * ce + GAMMA * region;
    }
}

// ---------------- launch ----------------
extern "C" void kernel_launch(void* const* d_in, const int* in_sizes, int n_in,
                              void* d_out, int out_size, void* d_ws, size_t ws_size,
                              hipStream_t stream) {
    const float* xin = (const float*)d_in[0];   // logits
    const float* tin = (const float*)d_in[1];   // targets
    float* out = (float*)d_out;

    char* ws = (char*)d_ws;
    float* partials = (float*)ws;                         // 4096*4 floats = 64 KB
    int*   counts   = (int*)(ws + 64 * 1024);             // 32 ints (pred | tgt)
    int*   labels   = (int*)(ws + 64 * 1024 + 256);       // 16 MB, reused pred->tgt
    int*   cnt_pred = counts;
    int*   cnt_tgt  = counts + 16;

    const int cclBlocks = NTOT / 256;                     // 16384

    hipLaunchKernelGGL(zero_counts_kernel, dim3(1), dim3(32), 0, stream, counts);
    hipLaunchKernelGGL(loss_reduce_kernel, dim3(RED_BLOCKS), dim3(RED_THREADS), 0, stream,
                       xin, tin, partials);

    // pred components: probs > 0.5  <=>  logits > 0
    hipLaunchKernelGGL(ccl_init_kernel,  dim3(cclBlocks), dim3(256), 0, stream, xin, 0.0f, labels);
    hipLaunchKernelGGL(ccl_merge_kernel, dim3(cclBlocks), dim3(256), 0, stream, labels);
    hipLaunchKernelGGL(ccl_count_kernel, dim3(cclBlocks), dim3(256), 0, stream, labels, cnt_pred);

    // target components: t > 0.5
    hipLaunchKernelGGL(ccl_init_kernel,  dim3(cclBlocks), dim3(256), 0, stream, tin, 0.5f, labels);
    hipLaunchKernelGGL(ccl_merge_kernel, dim3(cclBlocks), dim3(256), 0, stream, labels);
    hipLaunchKernelGGL(ccl_count_kernel, dim3(cclBlocks), dim3(256), 0, stream, labels, cnt_tgt);

    hipLaunchKernelGGL(finalize_kernel, dim3(1), dim3(32), 0, stream,
                       partials, cnt_pred, cnt_tgt, out);
}